// MyModel_61933428413143
// MI455X (gfx1250) — compile-verified
//
#include <hip/hip_runtime.h>

// Problem constants (reference: x is (32, 1, 1048576) fp32; a,b are 2-element fp32)
#define T_LEN      1048576
#define BATCH      32
#define CHUNK      4096            // samples per block
#define NTHREADS   256             // 8 wave32 per block
#define NWAVES     (NTHREADS / 32)
#define PER_THREAD 16              // CHUNK / NTHREADS
#define CPB        (T_LEN / CHUNK) // 256 chunks per row
#define NCHUNKS    (BATCH * CPB)   // 8192 blocks

typedef float v4f __attribute__((ext_vector_type(4)));

struct Coeffs { float c, b0, b1; };

__device__ __forceinline__ Coeffs load_coeffs(const float* __restrict__ a,
                                              const float* __restrict__ b) {
  float a0 = a[0];
  Coeffs k;
  k.c  = -a[1] / a0;
  k.b0 =  b[0] / a0;
  k.b1 =  b[1] / a0;
  return k;
}

// Affine pair (A,B) represents y_out = A*y_in + B.
// compose(p then q) = (q.A*p.A, q.A*p.B + q.B)

template <bool FINAL>
__global__ __launch_bounds__(NTHREADS)
void iir_chunk_kernel(const float* __restrict__ x,
                      const float* __restrict__ a,
                      const float* __restrict__ b,
                      float*       __restrict__ out,     // FINAL only
                      float*       __restrict__ aggAB,   // 2*NCHUNKS floats
                      const float* __restrict__ prefix)  // NCHUNKS floats (FINAL only)
{
  __shared__ __align__(16) float sx[CHUNK];  // 16 KB staged input
  __shared__ float swA[NWAVES];              // per-wave aggregates
  __shared__ float swB[NWAVES];

  const int t     = threadIdx.x;
  const int lane  = t & 31;
  const int wave  = t >> 5;
  const int blk   = blockIdx.x;
  const int chunk = blk % CPB;
  const long base = (long)(blk / CPB) * T_LEN + (long)chunk * CHUNK;
  const float* gsrc = x + base;

  // ---- CDNA5 async bulk copy: global -> LDS, 16B per lane per issue ----
  unsigned lds_base = (unsigned)(unsigned long long)(void*)sx;
#pragma unroll
  for (int j = 0; j < CHUNK / (NTHREADS * 4); ++j) {  // 4 issues/thread
    int fidx = (j * NTHREADS + t) * 4;                // float index, 16B granule
    const float* gp = gsrc + fidx;
    unsigned lp = lds_base + (unsigned)fidx * 4u;
    asm volatile("global_load_async_to_lds_b128 %0, %1, off"
                 :: "v"(lp), "v"(gp) : "memory");
  }
  asm volatile("s_wait_asynccnt 0x0" ::: "memory");   // own wave's copies done
  __syncthreads();                                    // all waves' copies visible

  const Coeffs k = load_coeffs(a, b);

  // Read this thread's 16 contiguous samples via ds_load_b128.
  const int local0 = t * PER_THREAD;
  float xs[PER_THREAD];
  {
    const v4f* sx4 = (const v4f*)sx;
#pragma unroll
    for (int i = 0; i < PER_THREAD / 4; ++i) {
      v4f v = sx4[(local0 >> 2) + i];
      xs[4 * i + 0] = v.x; xs[4 * i + 1] = v.y;
      xs[4 * i + 2] = v.z; xs[4 * i + 3] = v.w;
    }
  }
  float xprev;
  if (local0 == 0) {
    xprev = (chunk == 0) ? 0.0f : gsrc[-1];
  } else {
    xprev = sx[local0 - 1];
  }

  // FIR: f[n] = b0*x[n] + b1*x[n-1]; fold 16 steps of (y -> c*y + f) into (A,B).
  float f[PER_THREAD];
  float A = 1.0f, B = 0.0f;
#pragma unroll
  for (int i = 0; i < PER_THREAD; ++i) {
    f[i] = k.b0 * xs[i] + k.b1 * xprev;
    xprev = xs[i];
    B = k.c * B + f[i];
    A *= k.c;
  }

  // ---- wave32 inclusive scan of affine pairs via shuffles (no barriers) ----
  float wA = A, wB = B;
#pragma unroll
  for (int s = 1; s < 32; s <<= 1) {
    float pA = __shfl_up(wA, s, 32);
    float pB = __shfl_up(wB, s, 32);
    float nA = wA * pA;
    float nB = wA * pB + wB;
    if (lane >= s) { wA = nA; wB = nB; }
  }

  if (lane == 31) { swA[wave] = wA; swB[wave] = wB; }  // wave totals
  __syncthreads();

  if (!FINAL) {
    if (t == 0) {
      float tA = 1.0f, tB = 0.0f;   // compose wave totals in order
#pragma unroll
      for (int w = 0; w < NWAVES; ++w) {
        float qA = swA[w], qB = swB[w];
        tB = qA * tB + qB;
        tA = qA * tA;
      }
      aggAB[2 * blk]     = tA;
      aggAB[2 * blk + 1] = tB;
    }
    return;
  } else {
    // Exclusive prefix of earlier waves (<=7 LDS broadcast reads, uniform per wave).
    float wpA = 1.0f, wpB = 0.0f;
    for (int w = 0; w < wave; ++w) {
      float qA = swA[w], qB = swB[w];
      wpB = qA * wpB + qB;
      wpA = qA * wpA;
    }
    // Exclusive within wave: shift inclusive scan down by one lane.
    float eA = __shfl_up(wA, 1, 32);
    float eB = __shfl_up(wB, 1, 32);
    if (lane == 0) { eA = 1.0f; eB = 0.0f; }
    // Thread-exclusive total: wave-prefix first, then in-wave exclusive.
    float exA = eA * wpA;
    float exB = eA * wpB + eB;

    float carry = prefix[blk];        // chunk-incoming y
    float y = exA * carry + exB;      // thread-incoming y

    float o[PER_THREAD];
#pragma unroll
    for (int i = 0; i < PER_THREAD; ++i) {
      y = k.c * y + f[i];
      o[i] = fminf(1.0f, fmaxf(-1.0f, y));   // clamp=True
    }

    // Non-temporal 128-bit stores: keep x resident in the 192MB L2 for the
    // concurrent re-read pass (x = 128MB fits; writes shouldn't evict it).
    v4f* dst = (v4f*)(out + base + local0);
#pragma unroll
    for (int i = 0; i < PER_THREAD / 4; ++i) {
      v4f v = { o[4 * i], o[4 * i + 1], o[4 * i + 2], o[4 * i + 3] };
      __builtin_nontemporal_store(v, dst + i);
    }
  }
}

// Phase 2: per-row serial scan over 256 chunk aggregates (tiny, one wave).
__global__ __launch_bounds__(32)
void iir_carry_kernel(const float* __restrict__ aggAB,
                      float*       __restrict__ prefix)
{
  int row = threadIdx.x;
  if (row >= BATCH) return;
  const float* ab = aggAB + (size_t)row * CPB * 2;
  float* pf = prefix + (size_t)row * CPB;
  float carry = 0.0f;
  for (int i = 0; i < CPB; ++i) {
    pf[i] = carry;
    carry = ab[2 * i] * carry + ab[2 * i + 1];
  }
}

extern "C" void kernel_launch(void* const* d_in, const int* in_sizes, int n_in,
                              void* d_out, int out_size, void* d_ws, size_t ws_size,
                              hipStream_t stream) {
  (void)in_sizes; (void)n_in; (void)out_size; (void)ws_size;
  const float* x = (const float*)d_in[0];
  const float* a = (const float*)d_in[1];
  const float* b = (const float*)d_in[2];
  float* out = (float*)d_out;

  // Workspace: [0, 2*NCHUNKS) chunk (A,B) pairs; [2*NCHUNKS, 3*NCHUNKS) prefixes. 96 KB.
  float* aggAB  = (float*)d_ws;
  float* prefix = aggAB + 2 * NCHUNKS;

  iir_chunk_kernel<false><<<NCHUNKS, NTHREADS, 0, stream>>>(x, a, b, nullptr, aggAB, nullptr);
  iir_carry_kernel<<<1, 32, 0, stream>>>(aggAB, prefix);
  iir_chunk_kernel<true><<<NCHUNKS, NTHREADS, 0, stream>>>(x, a, b, out, aggAB, prefix);
}